// OrderedFlowModel_24232205484526
// MI455X (gfx1250) — compile-verified
//
#include <hip/hip_runtime.h>
#include <math.h>

typedef __attribute__((ext_vector_type(2))) float v2f;
typedef __attribute__((ext_vector_type(8))) float v8f;
typedef int vint4 __attribute__((vector_size(4 * sizeof(int))));  // matches builtin param

#define P_DIM      128
#define ROWS       64          // rows per block
#define BLOCK      128         // 4 wave32
#define LDS_STRIDE 132         // 128 + 4-float pad: 16 A-rows land on distinct bank pairs

// parameter block layout in d_ws (floats)
#define OFF_A     0
#define OFF_B     1
#define OFF_ALPHA 2    // 4 cutpoints
#define OFF_CW    6    // 17 input-side knots (outputs of inverse)
#define OFF_CH    23   // 17 output-side knots (inputs of inverse)
#define OFF_D     40   // 17 derivatives
#define OFF_W     57   // 16 bin widths
#define OFF_H     73   // 16 bin heights
#define NPARAM    96

#define LOG_HALF  (-0.69314718f)
#define INV_SQRT2 (0.70710678f)

// ---- CDNA5 async memory->LDS path (guarded; falls back to VGPR staging) ----
#if defined(__has_builtin)
#if __has_builtin(__builtin_amdgcn_global_load_async_to_lds_b128) && \
    __has_builtin(__builtin_amdgcn_s_wait_asynccnt)
#define USE_ASYNC_LDS 1
#endif
#endif
#ifndef USE_ASYNC_LDS
#define USE_ASYNC_LDS 0
#endif

__device__ __forceinline__ float softplusf(float x) {
  return x > 20.f ? x : log1pf(expf(x));
}

__device__ void knots_dev(const float* u, float* c) {  // c[17]
  float m = u[0];
  for (int i = 1; i < 16; ++i) m = fmaxf(m, u[i]);
  float e[16];
  float s = 0.f;
  for (int i = 0; i < 16; ++i) { e[i] = expf(u[i] - m); s += e[i]; }
  float cum = 0.f;
  c[0] = -8.f;
  for (int i = 0; i < 16; ++i) {
    float w = 1e-3f + (1.f - 1e-3f * 16.f) * (e[i] / s);
    cum += w;
    c[i + 1] = -8.f + 16.f * cum;
  }
  c[16] = 8.f;
}

__global__ void prep_kernel(const float* alpha1, const float* gaps_raw,
                            const float* uw, const float* uh, const float* ud,
                            const float* a_raw, const float* b,
                            float* ws, float* out) {
  if (threadIdx.x != 0 || blockIdx.x != 0) return;
  out[0] = 0.f;                                  // d_out is poisoned; re-zero every call
  ws[OFF_A] = softplusf(a_raw[0]);
  ws[OFF_B] = b[0];
  float acc = alpha1[0];
  ws[OFF_ALPHA + 0] = acc;
  for (int j = 0; j < 3; ++j) {                  // J-2 = 3 gaps -> 4 cutpoints
    acc += softplusf(gaps_raw[j]) + 1e-4f;
    ws[OFF_ALPHA + 1 + j] = acc;
  }
  knots_dev(uw, ws + OFF_CW);
  knots_dev(uh, ws + OFF_CH);
  for (int i = 0; i < 17; ++i) ws[OFF_D + i] = 1e-3f + softplusf(ud[i]);
  for (int i = 0; i < 16; ++i) ws[OFF_W + i] = ws[OFF_CW + i + 1] - ws[OFF_CW + i];
  for (int i = 0; i < 16; ++i) ws[OFF_H + i] = ws[OFF_CH + i + 1] - ws[OFF_CH + i];
}

__device__ __forceinline__ float spline_inv(float x, const float* Pm) {
  const float* cw = Pm + OFF_CW;
  const float* ch = Pm + OFF_CH;
  const float* dd = Pm + OFF_D;
  const float* w  = Pm + OFF_W;
  const float* h  = Pm + OFF_H;
  bool inside = (x >= -8.f) && (x <= 8.f);
  float xc = fminf(fmaxf(x, -8.f), 8.f);
  int idx = 0;
#pragma unroll
  for (int t = 1; t <= 16; ++t) idx += (xc >= ch[t]) ? 1 : 0;
  idx = min(idx, 15);
  float chk = ch[idx], hk = h[idx];
  float cwk = cw[idx], wk = w[idx];
  float dk = dd[idx], dk1 = dd[idx + 1];
  float delta = hk / wk;
  float s  = dk + dk1 - 2.f * delta;
  float dx = xc - chk;
  float aq = dx * s + hk * (delta - dk);
  float bq = hk * dk - dx * s;
  float cq = -delta * dx;
  float disc = fmaxf(bq * bq - 4.f * aq * cq, 0.f);
  float root = 2.f * cq / (-bq - sqrtf(disc));
  float z = root * wk + cwk;
  return inside ? z : x;
}

__device__ __forceinline__ float log_ndtr_f(float z) {
  return LOG_HALF + logf(erfcf(-z * INV_SQRT2));
}
__device__ __forceinline__ float log_surv_f(float z) {
  return LOG_HALF + logf(erfcf(z * INV_SQRT2));
}

__global__ void __launch_bounds__(BLOCK)
ordered_flow_nll_kernel(const float* __restrict__ X, const int* __restrict__ y,
                        const float* __restrict__ beta, const float* __restrict__ ws,
                        float* __restrict__ out, int n) {
  __shared__ __align__(16) float sX[ROWS * LDS_STRIDE];
  __shared__ __align__(16) float sBeta[P_DIM];
  __shared__ float sParam[NPARAM];
  __shared__ float sEta[ROWS];
  __shared__ float sRed[BLOCK / 32];

  const int t = threadIdx.x;
  const int rowblk = blockIdx.x * ROWS;

  // ---- staging: 64x128 fp32 tile of X -> LDS ----
#if USE_ASYNC_LDS
  // CDNA5 path: GLOBAL_LOAD_ASYNC_TO_LDS_B128 writes LDS directly (ASYNCcnt),
  // no VGPR round-trip in the staging loop.
#pragma unroll
  for (int i = 0; i < (ROWS * P_DIM / 4) / BLOCK; ++i) {   // 16 b128 per thread
    int f = i * BLOCK + t;          // flat float4 index
    int r = f >> 5;                 // 32 float4 per row
    int c4 = f & 31;
    float* lp = &sX[r * LDS_STRIDE + c4 * 4];
    if (rowblk + r < n) {
      const float* gp = X + (size_t)(rowblk + r) * P_DIM + c4 * 4;
      __builtin_amdgcn_global_load_async_to_lds_b128(
          (__attribute__((address_space(1))) vint4*)gp,
          (__attribute__((address_space(3))) vint4*)lp,
          0, 0);
    } else {
      *reinterpret_cast<float4*>(lp) = make_float4(0.f, 0.f, 0.f, 0.f);
    }
  }
  __builtin_amdgcn_s_wait_asynccnt(0);
#else
#pragma unroll
  for (int i = 0; i < (ROWS * P_DIM / 4) / BLOCK; ++i) {   // 16 float4 per thread
    int f = i * BLOCK + t;          // flat float4 index
    int r = f >> 5;                 // 32 float4 per row
    int c4 = f & 31;
    float4 v = make_float4(0.f, 0.f, 0.f, 0.f);
    if (rowblk + r < n)
      v = *reinterpret_cast<const float4*>(X + (size_t)(rowblk + r) * P_DIM + c4 * 4);
    *reinterpret_cast<float4*>(&sX[r * LDS_STRIDE + c4 * 4]) = v;
  }
#endif
  if (t < 32)
    *reinterpret_cast<float4*>(&sBeta[t * 4]) =
        *reinterpret_cast<const float4*>(beta + t * 4);
  if (t < NPARAM) sParam[t] = ws[t];
  __syncthreads();

  // ---- eta via V_WMMA_F32_16X16X4_F32: 16 rows per wave, K accumulated over 128 ----
  const int wv   = t >> 5;         // wave id 0..3
  const int lane = t & 31;
  const int m    = lane & 15;      // A-matrix row within tile
  const int hbit = lane >> 4;      // lane-half selects K pair
  const int rbase = wv * 16;

  v8f acc = {};
#pragma unroll
  for (int k0 = 0; k0 < P_DIM; k0 += 4) {
    v2f af = *reinterpret_cast<const v2f*>(&sX[(rbase + m) * LDS_STRIDE + k0 + 2 * hbit]);
    v2f bf = *reinterpret_cast<const v2f*>(&sBeta[k0 + 2 * hbit]);  // beta replicated over N
    acc = __builtin_amdgcn_wmma_f32_16x16x4_f32(
        false, af, false, bf, (short)0, acc, false, false);
  }

  // column 0 of D == eta; lane 0 holds M=0..7, lane 16 holds M=8..15
  if (m == 0) {
    int base = rbase + 8 * hbit;
#pragma unroll
    for (int j = 0; j < 8; ++j) sEta[base + j] = acc[j];
  }
  __syncthreads();

  // ---- fused elementwise tail: one row per thread (threads 0..63) ----
  float local = 0.f;
  if (t < ROWS && (rowblk + t) < n) {
    const float a  = sParam[OFF_A];
    const float bb = sParam[OFF_B];
    const float eta = sEta[t];
    const int yv = y[rowblk + t];            // 1..5
    const bool lf = yv > 1;
    const bool uf = yv < 5;
    const float al = lf ? sParam[OFF_ALPHA + yv - 2] : 0.f;
    const float au = uf ? sParam[OFF_ALPHA + yv - 1] : 0.f;
    const float el = (al - eta - bb) / a;
    const float eu = (au - eta - bb) / a;
    const float zl_s = lf ? spline_inv(el, sParam) : -12.f;
    const float zu_s = uf ? spline_inv(eu, sParam) :  12.f;
    const bool left  = (uf && zu_s <= 0.f) && (!lf || zl_s <= 0.f);
    const bool right = (!uf || zu_s >= 0.f) && (lf && zl_s >= 0.f);
    float lp;
    if (left) {
      float a_l = log_ndtr_f(zu_s);
      float d_l = fminf(log_ndtr_f(zl_s) - a_l, -1e-7f);
      lp = a_l + log1pf(-expf(d_l));
    } else if (right) {
      float a_r = log_surv_f(zl_s);
      float d_r = fminf(log_surv_f(zu_s) - a_r, -1e-7f);
      lp = a_r + log1pf(-expf(d_r));
    } else {
      float p = 0.5f * erfcf(-zu_s * INV_SQRT2) - 0.5f * erfcf(-zl_s * INV_SQRT2);
      lp = logf(fmaxf(p, 1e-30f));
    }
    local = -lp;
  }

  // ---- reduction: shuffle within wave32, LDS across waves, one atomic per block ----
  float v = local;
#pragma unroll
  for (int off = 16; off > 0; off >>= 1) v += __shfl_down(v, off);
  if (lane == 0) sRed[wv] = v;
  __syncthreads();
  if (t == 0) {
    float s = 0.f;
#pragma unroll
    for (int i = 0; i < BLOCK / 32; ++i) s += sRed[i];
    atomicAdd(out, s);
  }
}

extern "C" void kernel_launch(void* const* d_in, const int* in_sizes, int n_in,
                              void* d_out, int out_size, void* d_ws, size_t ws_size,
                              hipStream_t stream) {
  const float* X        = (const float*)d_in[0];
  const int*   y        = (const int*)d_in[1];
  const float* beta     = (const float*)d_in[2];
  const float* alpha1   = (const float*)d_in[3];
  const float* gaps_raw = (const float*)d_in[4];
  const float* uw       = (const float*)d_in[5];
  const float* uh       = (const float*)d_in[6];
  const float* ud       = (const float*)d_in[7];
  const float* a_raw    = (const float*)d_in[8];
  const float* b        = (const float*)d_in[9];
  float* out = (float*)d_out;
  float* ws  = (float*)d_ws;
  const int n = in_sizes[1];

  hipLaunchKernelGGL(prep_kernel, dim3(1), dim3(32), 0, stream,
                     alpha1, gaps_raw, uw, uh, ud, a_raw, b, ws, out);

  const int nblk = (n + ROWS - 1) / ROWS;
  hipLaunchKernelGGL(ordered_flow_nll_kernel, dim3(nblk), dim3(BLOCK), 0, stream,
                     X, y, beta, ws, out, n);
}